// GCN_5884105195982
// MI455X (gfx1250) — compile-verified
//
#include <hip/hip_runtime.h>

// GCN: out = ( GCNConv2( relu(GCNConv1(x)) ) ) @ Wl + bl
// N=100000 nodes, E=1600000 edges, 128 -> 64 -> 64 -> 1

#define N_NODES 100000
#define N_EDGES 1600000
#define IN_DIM  128
#define HID     64

typedef __attribute__((ext_vector_type(2))) float v2f;
typedef __attribute__((ext_vector_type(8))) float v8f;

// ---------------- degree / normalization ----------------

__global__ __launch_bounds__(256) void k_init_deg(float* __restrict__ deg) {
  int i = blockIdx.x * 256 + threadIdx.x;
  if (i < N_NODES) deg[i] = 1.0f;  // self-loop
}

__global__ __launch_bounds__(256) void k_accum_deg(const long long* __restrict__ dst,
                                                   float* __restrict__ deg) {
  int e = blockIdx.x * 256 + threadIdx.x;
  if (e < N_EDGES) atomicAdd(&deg[dst[e]], 1.0f);
}

__global__ __launch_bounds__(256) void k_rsqrt(float* __restrict__ deg) {
  int i = blockIdx.x * 256 + threadIdx.x;
  if (i < N_NODES) deg[i] = 1.0f / sqrtf(deg[i]);
}

// ---------------- WMMA GEMM: Out[N,64] = A[N,K] @ W[K,64] ----------------
// block = 256 threads = 8 wave32 waves; block tile = 32 rows x 64 cols.
// Each wave computes one 16x16 tile with V_WMMA_F32_16X16X4_F32, K-steps of 4.
// A-frag per ISA 7.12.2: lanes 0-15 hold K{+0,+1}, lanes 16-31 hold K{+2,+3}.
// C/D: VGPR v -> row (v + 8*(lane>=16)), col = lane&15.

template <int K, bool RELU_IN>
__global__ __launch_bounds__(256) void k_gemm_wmma(const float* __restrict__ A,
                                                   const float* __restrict__ W,
                                                   float* __restrict__ Out) {
  __shared__ float Xs[32 * K];
  __shared__ float Ws[K * 64];
  const int tid = threadIdx.x;
  const long long rowbase = (long long)blockIdx.x * 32;

  // cooperative coalesced float4 staging
  const float4* Ag = (const float4*)(A + rowbase * K);
  float4* Xs4 = (float4*)Xs;
  for (int i = tid; i < (32 * K) / 4; i += 256) {
    float4 v = Ag[i];
    if (RELU_IN) {
      v.x = fmaxf(v.x, 0.0f); v.y = fmaxf(v.y, 0.0f);
      v.z = fmaxf(v.z, 0.0f); v.w = fmaxf(v.w, 0.0f);
    }
    Xs4[i] = v;
  }
  const float4* Wg = (const float4*)W;
  float4* Ws4 = (float4*)Ws;
  for (int i = tid; i < (K * 64) / 4; i += 256) Ws4[i] = Wg[i];
  __syncthreads();

  const int wave = tid >> 5;
  const int lane = tid & 31;
  const int half = lane >> 4;   // 0: lanes 0-15, 1: lanes 16-31
  const int l15  = lane & 15;
  const int rt   = wave >> 2;   // row tile 0..1
  const int ct   = wave & 3;    // col tile 0..3
  const int m    = rt * 16 + l15;
  const int col  = ct * 16 + l15;

  v8f acc = {};
  #pragma unroll
  for (int k = 0; k < K; k += 4) {
    v2f a, b;
    a[0] = Xs[m * K + k + 2 * half + 0];
    a[1] = Xs[m * K + k + 2 * half + 1];
    b[0] = Ws[(k + 2 * half + 0) * 64 + col];
    b[1] = Ws[(k + 2 * half + 1) * 64 + col];
    acc = __builtin_amdgcn_wmma_f32_16x16x4_f32(false, a, false, b,
                                                (short)0, acc, false, false);
  }

  long long obase = (rowbase + rt * 16 + half * 8) * 64 + col;
  #pragma unroll
  for (int v = 0; v < 8; ++v) Out[obase + (long long)v * 64] = acc[v];
}

// ---------------- aggregation ----------------

// agg[i][f] = b[f] + h[i][f] * dinv[i]^2   (bias + self-loop message)
__global__ __launch_bounds__(256) void k_init_agg(const float* __restrict__ h,
                                                  const float* __restrict__ dinv,
                                                  const float* __restrict__ b,
                                                  float* __restrict__ agg) {
  int i = blockIdx.x * 256 + threadIdx.x;  // over N*64, exact
  int node = i >> 6;
  int f = i & 63;
  float di = dinv[node];
  agg[i] = b[f] + h[i] * di * di;
}

// one wave per edge; each lane handles 2 features; L2-resident atomics
__global__ __launch_bounds__(256) void k_scatter(const long long* __restrict__ src,
                                                 const long long* __restrict__ dst,
                                                 const float* __restrict__ dinv,
                                                 const float* __restrict__ h,
                                                 float* __restrict__ agg) {
  int e = blockIdx.x * 8 + (threadIdx.x >> 5);  // E/8 blocks, exact
  int lane = threadIdx.x & 31;
  long long s = src[e];
  long long d = dst[e];
  float nrm = dinv[s] * dinv[d];
  const float* hr = h + s * 64;
  float* ar = agg + d * 64;
  atomicAdd(ar + lane,      hr[lane]      * nrm);
  atomicAdd(ar + lane + 32, hr[lane + 32] * nrm);
}

// ---------------- final linear: out[i] = agg[i,:] . Wl + bl ----------------

__global__ __launch_bounds__(256) void k_final(const float* __restrict__ agg,
                                               const float* __restrict__ Wl,
                                               const float* __restrict__ bl,
                                               float* __restrict__ out) {
  int node = blockIdx.x * 8 + (threadIdx.x >> 5);  // N/8 blocks, exact
  int lane = threadIdx.x & 31;
  long long base = (long long)node * 64;
  float v = agg[base + lane] * Wl[lane] + agg[base + lane + 32] * Wl[lane + 32];
  #pragma unroll
  for (int off = 16; off > 0; off >>= 1) v += __shfl_down(v, off, 32);
  if (lane == 0) out[node] = v + bl[0];
}

// ---------------- driver ----------------

extern "C" void kernel_launch(void* const* d_in, const int* in_sizes, int n_in,
                              void* d_out, int out_size, void* d_ws, size_t ws_size,
                              hipStream_t stream) {
  (void)in_sizes; (void)n_in; (void)out_size; (void)ws_size;
  const float*     x   = (const float*)d_in[0];
  const long long* ei  = (const long long*)d_in[1];  // int64 [2, E]
  const float*     W1  = (const float*)d_in[2];
  const float*     b1  = (const float*)d_in[3];
  const float*     W2  = (const float*)d_in[4];
  const float*     b2  = (const float*)d_in[5];
  const float*     Wl  = (const float*)d_in[6];
  const float*     bl  = (const float*)d_in[7];
  float* out = (float*)d_out;

  const long long* src = ei;
  const long long* dst = ei + N_EDGES;

  char* ws = (char*)d_ws;
  size_t degBytes = (((size_t)N_NODES * 4) + 255) & ~(size_t)255;
  float* deg    = (float*)ws;
  float* hbuf   = (float*)(ws + degBytes);            // N x 64
  float* aggbuf = hbuf + (size_t)N_NODES * HID;       // N x 64

  // degree + dinv
  k_init_deg <<<(N_NODES + 255) / 256, 256, 0, stream>>>(deg);
  k_accum_deg<<<N_EDGES / 256,         256, 0, stream>>>(dst, deg);
  k_rsqrt    <<<(N_NODES + 255) / 256, 256, 0, stream>>>(deg);

  // layer 1: h = x @ W1 ; agg = b1 + self + scatter
  k_gemm_wmma<IN_DIM, false><<<N_NODES / 32, 256, 0, stream>>>(x, W1, hbuf);
  k_init_agg<<<(N_NODES * HID) / 256, 256, 0, stream>>>(hbuf, deg, b1, aggbuf);
  k_scatter <<<N_EDGES / 8,           256, 0, stream>>>(src, dst, deg, hbuf, aggbuf);

  // layer 2: h = relu(agg) @ W2 ; agg = b2 + self + scatter  (buffers reused)
  k_gemm_wmma<HID, true><<<N_NODES / 32, 256, 0, stream>>>(aggbuf, W2, hbuf);
  k_init_agg<<<(N_NODES * HID) / 256, 256, 0, stream>>>(hbuf, deg, b2, aggbuf);
  k_scatter <<<N_EDGES / 8,           256, 0, stream>>>(src, dst, deg, hbuf, aggbuf);

  // readout
  k_final<<<N_NODES / 8, 256, 0, stream>>>(aggbuf, Wl, bl, out);
}